// Node2VecSkipGram_47949014892834
// MI455X (gfx1250) — compile-verified
//
#include <hip/hip_runtime.h>

#define EMB_DIM 128
#define N_NEG   20
#define N_PROD  (1 + N_NEG)
#define TILE    16
#define LSTRIDE 132   // padded LDS row stride in floats: bank(132*r + c) = 4r + c -> conflict-free
#define EPS     1e-9f

typedef float v2f __attribute__((ext_vector_type(2)));
typedef float v8f __attribute__((ext_vector_type(8)));

// -log(sigmoid(x) + eps)
__device__ __forceinline__ float neg_log_sigmoid(float x) {
    float e   = __expf(-x);
    float sig = __builtin_amdgcn_rcpf(1.0f + e);
    return -__logf(sig + EPS);
}

// One wave gathers one 512B embedding row straight into LDS: 32 lanes x b128.
// vdst = per-lane LDS byte address, vaddr = per-lane 64-bit global address.
__device__ __forceinline__ void async_row_to_lds(const float* gsrc, unsigned lds_byte_addr) {
    asm volatile("global_load_async_to_lds_b128 %0, %1, off"
                 :: "v"(lds_byte_addr), "v"(gsrc) : "memory");
}
__device__ __forceinline__ void wait_async_le16() {
    asm volatile("s_wait_asynccnt 0x10" ::: "memory");
}
__device__ __forceinline__ void wait_async_0() {
    asm volatile("s_wait_asynccnt 0x0" ::: "memory");
}
__device__ __forceinline__ void wait_ds_0() {
    asm volatile("s_wait_dscnt 0x0" ::: "memory");
}

__global__ void __launch_bounds__(32)
node2vec_sgns_wmma_kernel(const int* __restrict__ center_nodes,
                          const int* __restrict__ context_nodes,
                          const int* __restrict__ negative_nodes,
                          const float* __restrict__ center_w,
                          const float* __restrict__ context_w,
                          float* __restrict__ out,
                          float inv_batch, int batch)
{
    __shared__ float ldsA[TILE * LSTRIDE];        // center tile (16 x 128, padded)
    __shared__ float ldsB[2][TILE * LSTRIDE];     // double-buffered context/negative tiles

    const int lane = threadIdx.x;                 // 0..31, wave32
    const int b0   = blockIdx.x * TILE;
    const int row  = lane & 15;
    const int half = lane >> 4;

    const unsigned laneByte = (unsigned)(lane * 16);
    const unsigned aBase  = (unsigned)(size_t)&ldsA[0];          // flat low 32 bits == LDS offset
    const unsigned bBase0 = (unsigned)(size_t)&ldsB[0][0];
    const unsigned bBase1 = (unsigned)(size_t)&ldsB[1][0];

    // Issue 16 async row-gathers (one tile) for product `prodk` into LDS buffer at `lbase`.
    auto issue_B_tile = [&](int prodk, unsigned lbase) {
        for (int r = 0; r < TILE; ++r) {
            int bi = b0 + r; if (bi >= batch) bi = batch - 1;
            int node = (prodk == 0) ? context_nodes[bi]
                                    : negative_nodes[(size_t)bi * N_NEG + (prodk - 1)];
            async_row_to_lds(context_w + (size_t)node * EMB_DIM + 4 * lane,
                             lbase + (unsigned)(r * LSTRIDE * 4) + laneByte);
        }
    };

    // ---- prologue: A tile, then B tiles for prod 0 and 1 (48 async ops in flight) ----
    for (int r = 0; r < TILE; ++r) {
        int bi = b0 + r; if (bi >= batch) bi = batch - 1;
        int node = center_nodes[bi];
        async_row_to_lds(center_w + (size_t)node * EMB_DIM + 4 * lane,
                         aBase + (unsigned)(r * LSTRIDE * 4) + laneByte);
    }
    issue_B_tile(0, bBase0);
    issue_B_tile(1, bBase1);

    // lanes 0-7 hold diag rows m=lane (VGPR m); lanes 24-31 hold m=lane-16 (VGPR m-8)
    const bool diag_lane = (half == 0) ? (row < 8) : (row >= 8);
    const bool row_valid = diag_lane && (b0 + row < batch);
    float row_loss = 0.0f;

    for (int prod = 0; prod < N_PROD; ++prod) {
        // In-order async completion: <=16 outstanding ==> everything except the
        // most recently issued tile has landed, so buf[prod&1] (and A) are ready.
        if (prod == N_PROD - 1) wait_async_0();
        else                    wait_async_le16();

        const float* Bbuf = (prod & 1) ? ldsB[1] : ldsB[0];

        // ---- 16x16x128 via 32 chained V_WMMA_F32_16X16X4_F32 ----
        // A frag (16x4): lane L -> row L&15, cols {4s+2h, 4s+2h+1}
        // B frag (4x16): lane L -> col  L&15, k    {2h, 2h+1}  -> same LDS offsets
        v8f acc = {};
        #pragma unroll
        for (int s = 0; s < EMB_DIM / 4; ++s) {
            int off = row * LSTRIDE + 4 * s + 2 * half;
            v2f a = *(const v2f*)(&ldsA[off]);
            v2f b = *(const v2f*)(&Bbuf[off]);
            acc = __builtin_amdgcn_wmma_f32_16x16x4_f32(
                      /*neg_a=*/false, a, /*neg_b=*/false, b,
                      /*c_mod=*/(short)0, acc, /*reuse_a=*/false, /*reuse_b=*/false);
        }

        // ---- refill the buffer we just consumed with tile prod+2 ----
        if (prod + 2 < N_PROD) {
            wait_ds_0();   // WAR: all ds_loads of this buffer have returned
            issue_B_tile(prod + 2, (prod & 1) ? bBase1 : bBase0);
        }

        // ---- extract this lane's diagonal element (score for batch row b0+row) ----
        int sel = row & 7;
        float sc = acc[0];
        sc = (sel == 1) ? acc[1] : sc;
        sc = (sel == 2) ? acc[2] : sc;
        sc = (sel == 3) ? acc[3] : sc;
        sc = (sel == 4) ? acc[4] : sc;
        sc = (sel == 5) ? acc[5] : sc;
        sc = (sel == 6) ? acc[6] : sc;
        sc = (sel == 7) ? acc[7] : sc;

        float contrib = (prod == 0) ? neg_log_sigmoid(sc)     // positive pair
                                    : neg_log_sigmoid(-sc);   // negative pair
        row_loss += row_valid ? contrib : 0.0f;
    }

    // ---- wave32 sum of the 16 per-row losses, then one atomic into the mean ----
    float total = row_loss;
    #pragma unroll
    for (int m = 16; m >= 1; m >>= 1) total += __shfl_xor(total, m, 32);
    if (lane == 0) atomicAdd(out, total * inv_batch);
}

extern "C" void kernel_launch(void* const* d_in, const int* in_sizes, int n_in,
                              void* d_out, int out_size, void* d_ws, size_t ws_size,
                              hipStream_t stream) {
    const int*   center_nodes   = (const int*)d_in[0];
    const int*   context_nodes  = (const int*)d_in[1];
    const int*   negative_nodes = (const int*)d_in[2];
    const float* center_w       = (const float*)d_in[3];
    const float* context_w      = (const float*)d_in[4];
    float*       out            = (float*)d_out;

    const int batch = in_sizes[0];
    const int nblocks = (batch + TILE - 1) / TILE;   // 65536/16 = 4096 tiles

    hipMemsetAsync(d_out, 0, sizeof(float), stream); // graph-capture legal
    node2vec_sgns_wmma_kernel<<<nblocks, 32, 0, stream>>>(
        center_nodes, context_nodes, negative_nodes,
        center_w, context_w, out, 1.0f / (float)batch, batch);
}